// FullyConnectedTEEncoder_27135603376523
// MI455X (gfx1250) — compile-verified
//
#include <hip/hip_runtime.h>
#include <hip/hip_bf16.h>

// r[b,n,:] = (mean_m relu(U[b,m,:] + V[b,n,:])) @ W2 + b2
//   U[b,m,h] =  x_ctx[b,m]·W1[0:2,h] + y_ctx[b,m]·W1[2:4,h]
//   V[b,n,h] = -x_ctx[b,n]·W1[0:2,h] + y_ctx[b,n]·W1[4:6,h] + b1[h]
//
// Block = 256 threads (8 wave32) handles one (b, 16-row n-tile).
// Each wave: V tile via V_WMMA_F32_16X16X4_F32, then accumulates
// relu(U+V) over its 128-m slice. LDS ds_add_f32 reduction across waves,
// then wave 0 projects through W2 with 4 chained K=4 f32 WMMAs.

#define BB   8
#define NN   1024
#define HH   16
#define OO   16

typedef __attribute__((ext_vector_type(2))) float v2f;
typedef __attribute__((ext_vector_type(8))) float v8f;

__global__ __launch_bounds__(256) void te_encoder_kernel(
    const float* __restrict__ x_ctx,   // [B,N,2]
    const float* __restrict__ y_ctx,   // [B,N,2]
    const float* __restrict__ W1,      // [6,16] row-major
    const float* __restrict__ b1,      // [16]
    const float* __restrict__ W2,      // [16,16] row-major
    const float* __restrict__ b2,      // [16]
    float* __restrict__ out)           // [B,N,16]
{
    __shared__ float sred[16 * 16];    // S[n][h], row-major within the tile

    const int tid  = threadIdx.x;
    const int lane = tid & 31;
    const int wave = __builtin_amdgcn_readfirstlane(tid >> 5);  // 0..7, wave-uniform

    const int tile = blockIdx.x;        // 0..511
    const int b    = tile >> 6;         // N/16 = 64 tiles per batch
    const int n0   = (tile & 63) << 4;

    const int  hcol = lane & 15;        // "N" column index for this lane (h)
    const bool hi   = lane >= 16;
    const int  nbase = hi ? 8 : 0;      // C/D layout: lanes>=16 hold rows M=8..15

    // ---------------- V tile via WMMA f32 16x16x4 ----------------
    // A (16x4 f32, 2 VGPRs): lanes 0-15  -> A[M=lane, K=0..1] = (-x0,-x1) of row n0+lane
    //                        lanes 16-31 -> A[M=lane-16, K=2..3] = (y0,y1) of row n0+lane-16
    const int nrow = n0 + hcol;
    v2f av;
    if (!hi) {
        av.x = -x_ctx[(b * NN + nrow) * 2 + 0];
        av.y = -x_ctx[(b * NN + nrow) * 2 + 1];
    } else {
        av.x =  y_ctx[(b * NN + nrow) * 2 + 0];
        av.y =  y_ctx[(b * NN + nrow) * 2 + 1];
    }
    // B (4x16 f32, 2 VGPRs, mirrored layout): VGPR0: K=0 (lanes 0-15) / K=2 (lanes 16-31)
    //                                         VGPR1: K=1              / K=3
    // rows K=0..3 are W1 rows {0,1,4,5} (x-diff part, y_n part)
    v2f bv;
    if (!hi) { bv.x = W1[0 * 16 + hcol]; bv.y = W1[1 * 16 + hcol]; }
    else     { bv.x = W1[4 * 16 + hcol]; bv.y = W1[5 * 16 + hcol]; }
    // C = b1 broadcast: every (n,h) cell gets b1[h]
    const float bias1 = b1[hcol];
    v8f cin;
#pragma unroll
    for (int i = 0; i < 8; ++i) cin[i] = bias1;

    // D layout: VGPR r, lane<16 -> V[n=r, h=lane]; lane>=16 -> V[n=8+r, h=lane-16]
    v8f V = __builtin_amdgcn_wmma_f32_16x16x4_f32(
        false, av, false, bv, (short)0, cin, false, false);

    // ---------------- accumulate S = sum_m relu(U[m] + V) ----------------
    // Per-lane W1 columns for U (rows 0,1 = x part; rows 2,3 = y_m part)
    const float wu0 = W1[0 * 16 + hcol];
    const float wu1 = W1[1 * 16 + hcol];
    const float wu2 = W1[2 * 16 + hcol];
    const float wu3 = W1[3 * 16 + hcol];

    float S[8];
#pragma unroll
    for (int r = 0; r < 8; ++r) S[r] = 0.0f;

    const float* xb = x_ctx + b * NN * 2;
    const float* yb = y_ctx + b * NN * 2;

    const int m0 = wave * (NN / 8);
    const int m1 = m0 + (NN / 8);
    for (int m = m0; m < m1; ++m) {
        // wave-uniform scalars -> scalar loads / broadcast
        const float xm0 = xb[2 * m + 0];
        const float xm1 = xb[2 * m + 1];
        const float ym0 = yb[2 * m + 0];
        const float ym1 = yb[2 * m + 1];
        const float u = xm0 * wu0 + xm1 * wu1 + ym0 * wu2 + ym1 * wu3;
#pragma unroll
        for (int r = 0; r < 8; ++r) {
            const float z = u + V[r];
            S[r] += fmaxf(z, 0.0f);
        }
    }

    // ---------------- cross-wave reduction into LDS (row-major S[n][h]) ----------------
    sred[tid] = 0.0f;                // 256 threads zero all 256 cells
    __syncthreads();
#pragma unroll
    for (int r = 0; r < 8; ++r) {
        atomicAdd(&sred[(nbase + r) * 16 + hcol], S[r]);   // ds_add_f32
    }
    __syncthreads();

    // ---------------- wave 0: R = (S/N) @ W2 + b2 via 4 chained K=4 WMMAs ----------------
    if (tid < 32) {
        const float inv = 1.0f / (float)NN;
        const float bias2 = b2[hcol];
        v8f acc;
#pragma unroll
        for (int i = 0; i < 8; ++i) acc[i] = bias2;

#pragma unroll
        for (int k = 0; k < 16; k += 4) {
            // A = Smean[:, k:k+4] in 16x4 A layout
            v2f a2, b2v;
            if (!hi) {
                a2.x  = sred[hcol * 16 + k + 0] * inv;
                a2.y  = sred[hcol * 16 + k + 1] * inv;
                b2v.x = W2[(k + 0) * 16 + hcol];
                b2v.y = W2[(k + 1) * 16 + hcol];
            } else {
                a2.x  = sred[hcol * 16 + k + 2] * inv;
                a2.y  = sred[hcol * 16 + k + 3] * inv;
                b2v.x = W2[(k + 2) * 16 + hcol];
                b2v.y = W2[(k + 3) * 16 + hcol];
            }
            acc = __builtin_amdgcn_wmma_f32_16x16x4_f32(
                false, a2, false, b2v, (short)0, acc, false, false);
        }

        // D layout -> out[b, n0+nbase+r, hcol]
#pragma unroll
        for (int r = 0; r < 8; ++r) {
            out[(b * NN + n0 + nbase + r) * OO + hcol] = acc[r];
        }
    }
}

extern "C" void kernel_launch(void* const* d_in, const int* in_sizes, int n_in,
                              void* d_out, int out_size, void* d_ws, size_t ws_size,
                              hipStream_t stream) {
    // setup_inputs() order: x_ctx, y_ctx, x_trg(unused), W1, b1, W2, b2
    const float* x_ctx = (const float*)d_in[0];
    const float* y_ctx = (const float*)d_in[1];
    const float* W1    = (const float*)d_in[3];
    const float* b1    = (const float*)d_in[4];
    const float* W2    = (const float*)d_in[5];
    const float* b2    = (const float*)d_in[6];
    float* out = (float*)d_out;

    // 512 blocks = 8 batches x 64 n-tiles; 256 threads = 8 wave32
    te_encoder_kernel<<<dim3(BB * (NN / 16)), dim3(256), 0, stream>>>(
        x_ctx, y_ctx, W1, b1, W2, b2, out);
}